// CornerBoundingBoxEMDLoss_20143396618917
// MI455X (gfx1250) — compile-verified
//
#include <hip/hip_runtime.h>
#include <math.h>

typedef __attribute__((ext_vector_type(2))) float v2f;
typedef __attribute__((ext_vector_type(8))) float v8f;

#define NB_PER_BLOCK 4   // batches per block (2 waves x 2 batches/wave)
#define THREADS      64

// Masks 0..255 ordered by popcount, plus level offsets (levels 0..8).
struct MaskTable {
  unsigned char masks[256];
  short off[10];
  constexpr MaskTable() : masks{}, off{} {
    int idx = 0;
    for (int k = 0; k <= 8; ++k) {
      off[k] = (short)idx;
      for (int m = 0; m < 256; ++m) {
        int pc = 0;
        for (int b = 0; b < 8; ++b) pc += (m >> b) & 1;
        if (pc == k) masks[idx++] = (unsigned char)m;
      }
    }
    off[9] = (short)idx;
  }
};
__constant__ MaskTable MT{};

__global__ __launch_bounds__(THREADS)
void CornerBoundingBoxEMDLoss_kernel(const float* __restrict__ pred,
                                     const float* __restrict__ tgt,
                                     float* __restrict__ out, int B) {
  __shared__ float distS[NB_PER_BLOCK][8][8];   // per-batch 8x8 distance
  __shared__ float dpS[NB_PER_BLOCK][256];      // per-batch subset DP

  const int tid   = threadIdx.x;
  const int lane  = tid & 31;
  const int wave  = tid >> 5;       // 0..1
  const int khalf = lane >> 4;      // 0: K=0,1  1: K=2,3 (half-wave K split)
  const int r     = lane & 15;      // M (for A) and N (for B) index

  // ---------------- Phase 1: dist^2 via V_WMMA_F32_16X16X4_F32 -------------
  // Rows M: 0-7 = batch bA pred corners, 8-15 = batch bA+1 pred corners.
  // Cols N: 0-7 = batch bA targets,      8-15 = batch bA+1 targets.
  const int bA = blockIdx.x * NB_PER_BLOCK + wave * 2;

  int mb = bA + (r >> 3); if (mb >= B) mb = B - 1;     // batch of row/col r
  const int mc = r & 7;                                // corner index

  // A fragment (16x4 f32, 2 VGPRs): lane holds A[M=r][K=2*khalf +0/+1]
  const float* P = pred + (size_t)mb * 24 + mc * 3;
  const float px = P[0], py = P[1], pz = P[2];
  v2f a;
  a.x = khalf ? (-2.0f * pz) : (-2.0f * px);
  a.y = khalf ? 1.0f         : (-2.0f * py);

  // B fragment (4x16 f32, 2 VGPRs): lane holds B[K=2*khalf +0/+1][N=r]
  const float* T = tgt + (size_t)mb * 24 + mc * 3;
  const float tx = T[0], ty = T[1], tz = T[2];
  const float tn = tx * tx + ty * ty + tz * tz;
  v2f bf;
  bf.x = khalf ? tz : tx;
  bf.y = khalf ? tn : ty;

  // C accumulator: C[M][N] = |p_M|^2. Lanes 0-15 -> M=v, lanes 16-31 -> M=8+v.
  int bh = bA + khalf; if (bh >= B) bh = B - 1;
  const float* Pc = pred + (size_t)bh * 24;
  v8f c;
#pragma unroll
  for (int v = 0; v < 8; ++v) {
    const float x = Pc[v * 3 + 0], y = Pc[v * 3 + 1], z = Pc[v * 3 + 2];
    c[v] = x * x + y * y + z * z;
  }

  // D[M][N] = |p_M|^2 - 2 p_M . t_N + |t_N|^2  (EXEC all-ones here)
  v8f d = __builtin_amdgcn_wmma_f32_16x16x4_f32(false, a, false, bf,
                                                (short)0, c, false, false);

  // D layout: lanes 0-15 -> (M=v, N=lane); lanes 16-31 -> (M=8+v, N=lane-16).
  // Batch0 block: lanes 0-7 (cols 0-7). Batch1 block: lanes 24-31 (cols 8-15).
  int q = -1, col = 0;
  if (lane < 8)        { q = wave * 2;     col = lane;      }
  else if (lane >= 24) { q = wave * 2 + 1; col = lane - 24; }
  if (q >= 0) {
#pragma unroll
    for (int v = 0; v < 8; ++v)
      distS[q][v][col] = __builtin_amdgcn_sqrtf(fmaxf(d[v], 0.0f));
  }

  // ---------------- Phase 2: Held-Karp assignment DP -----------------------
  // 16 lanes per batch; dp[mask] = min cost matching rows 0..k-1 to targets
  // in mask (popcount(mask)=k). Exact equivalent of min over 8! permutations.
  const int g   = tid >> 4;   // batch-in-block 0..3
  const int sub = tid & 15;
  if (sub == 0) dpS[g][0] = 0.0f;
  __syncthreads();

  for (int k = 1; k <= 8; ++k) {
    const int s = MT.off[k], e = MT.off[k + 1];
    const int i = k - 1;                 // pred row assigned at this level
    for (int t = s + sub; t < e; t += 16) {
      const int mask = MT.masks[t];
      float best = 3.4e38f;
#pragma unroll
      for (int j = 0; j < 8; ++j) {
        if (mask & (1 << j)) {
          const float v = dpS[g][mask ^ (1 << j)] + distS[g][i][j];
          best = fminf(best, v);
        }
      }
      dpS[g][mask] = best;
    }
    __syncthreads();   // level k+1 reads level k results (cross-wave safe)
  }

  const int batch = blockIdx.x * NB_PER_BLOCK + g;
  if (sub == 0 && batch < B) out[batch] = dpS[g][255];
}

extern "C" void kernel_launch(void* const* d_in, const int* in_sizes, int n_in,
                              void* d_out, int out_size, void* d_ws, size_t ws_size,
                              hipStream_t stream) {
  const float* pred = (const float*)d_in[0];   // [B,8,3] f32
  const float* tgt  = (const float*)d_in[1];   // [B,8,3] f32
  float* out = (float*)d_out;                  // [B] f32
  const int B = in_sizes[0] / 24;
  const int blocks = (B + NB_PER_BLOCK - 1) / NB_PER_BLOCK;
  CornerBoundingBoxEMDLoss_kernel<<<blocks, THREADS, 0, stream>>>(pred, tgt, out, B);
}